// AnovaKernel_86517821214540
// MI455X (gfx1250) — compile-verified
//
#include <hip/hip_runtime.h>

// ANOVA kernel DP (order 3): out[b] = sum_d e3(x[b, :, d]),  x: (B, 40, 64) fp32.
// HBM-bound single-pass stream (AI ~1.5 FLOP/byte, ~7.2us floor @ 23.3 TB/s).
// CDNA5 path: double-buffered GLOBAL_LOAD_ASYNC_TO_LDS_B128 staging (ASYNCcnt),
// consume via ds_load_b128, DP recurrence in 12 VGPRs, wave32 shuffle reduce.

typedef __attribute__((ext_vector_type(4))) float v4f;

constexpr int F = 40;
constexpr int D = 64;
constexpr int CH = 8;              // field rows per async chunk
constexpr int NCH = F / CH;        // 5 chunks
constexpr int THREADS = 256;       // 8 waves, 16 b-rows per block

__global__ __launch_bounds__(THREADS) void AnovaE3Async(
    const float* __restrict__ x, float* __restrict__ out, int B) {
  // Two 32 KB staging buffers: per thread CH consecutive v4f slots per buffer.
  __shared__ v4f sbuf[2][THREADS * CH];

  const int tid = threadIdx.x;
  const int lane = tid & 31;                         // wave32
  const int gwave = (blockIdx.x * THREADS + tid) >> 5;
  const int half = lane >> 4;                        // which b this half-wave owns
  const int sub  = lane & 15;
  const int b = gwave * 2 + half;
  if (b >= B) return;

  // Byte offset of this thread's slice of field row 0, relative to x.
  const unsigned row_bytes = D * 4u;
  unsigned base_off = ((unsigned)b * (unsigned)(F * D) + (unsigned)sub * 4u) * 4u;
  const int slot = tid * CH;

  // Issue one chunk (CH field rows) of async b128 global->LDS transfers.
  auto issue_chunk = [&](int c, int sel) {
    unsigned voff = base_off + (unsigned)(c * CH) * row_bytes;
#pragma unroll
    for (int fi = 0; fi < CH; ++fi) {
      // Flat->LDS: low 32 bits of the flat address are the LDS byte offset.
      unsigned lds = (unsigned)(uintptr_t)&sbuf[sel][slot + fi];
      asm volatile(
          "global_load_async_to_lds_b128 %0, %1, %2 th:TH_LOAD_NT"
          :: "v"(lds), "v"(voff), "s"(x)
          : "memory");
      voff += row_bytes;
    }
  };

  issue_chunk(0, 0);

  v4f a1 = {0.f, 0.f, 0.f, 0.f};
  v4f a2 = {0.f, 0.f, 0.f, 0.f};
  v4f a3 = {0.f, 0.f, 0.f, 0.f};

#pragma unroll
  for (int c = 0; c < NCH; ++c) {
    if (c + 1 < NCH) {
      // Buffer (c+1)&1 was last read in iteration c-1; make sure those
      // ds_loads retired before the async engine overwrites it.
      asm volatile("s_wait_dscnt 0x0" ::: "memory");
      issue_chunk(c + 1, (c + 1) & 1);
      // Async loads retire in order: <=8 outstanding => chunk c has landed.
      asm volatile("s_wait_asynccnt 0x8" ::: "memory");
    } else {
      asm volatile("s_wait_asynccnt 0x0" ::: "memory");
    }

    const v4f* __restrict__ buf = &sbuf[c & 1][slot];
#pragma unroll
    for (int fi = 0; fi < CH; ++fi) {
      v4f v = buf[fi];                 // ds_load_b128
      // Elementary-symmetric DP; order matters (pre-update a2, then a1).
      a3 += v * a2;
      a2 += v * a1;
      a1 += v;
    }
  }

  // Horizontal sum of 4 components, then across the 16 lanes of this half.
  float s = a3.x + a3.y + a3.z + a3.w;
  s += __shfl_down(s, 8, 16);
  s += __shfl_down(s, 4, 16);
  s += __shfl_down(s, 2, 16);
  s += __shfl_down(s, 1, 16);

  if (sub == 0) out[b] = s;
}

extern "C" void kernel_launch(void* const* d_in, const int* in_sizes, int n_in,
                              void* d_out, int out_size, void* d_ws, size_t ws_size,
                              hipStream_t stream) {
  const float* x = (const float*)d_in[0];
  float* out = (float*)d_out;
  const int B = in_sizes[0] / (F * D);   // 16384 for the reference shapes

  const int rows_per_block = (THREADS / 32) * 2;   // 16
  dim3 grid((B + rows_per_block - 1) / rows_per_block);
  AnovaE3Async<<<grid, THREADS, 0, stream>>>(x, out, B);
}